// GCN_81973745811883
// MI455X (gfx1250) — compile-verified
//
#include <hip/hip_runtime.h>

#define NNODES 100000
#define NEDGES 3200000
#define DIN    128
#define DH     256
#define DOUT   40
#define BN_EPS 1e-5f
#define NBN    256          // blocks used for deterministic BN partial reduction
#define NTILES (NNODES/16)  // 6250 exact

typedef float v2f __attribute__((ext_vector_type(2)));
typedef float v8f __attribute__((ext_vector_type(8)));

__device__ __forceinline__ v8f wmma_f32(v2f a, v2f b, v8f c) {
  // D = A(16x4,f32) * B(4x16,f32) + C(16x16,f32)
  return __builtin_amdgcn_wmma_f32_16x16x4_f32(false, a, false, b, (short)0, c,
                                               false, false);
}

// ---------------- degree / dinv ----------------
__global__ void init_deg(float* deg) {
  int n = blockIdx.x * blockDim.x + threadIdx.x;
  if (n < NNODES) deg[n] = 1.0f;  // self loop contributes 1
}

__global__ void deg_accum(const long long* __restrict__ ei, float* deg) {
  int stride = gridDim.x * blockDim.x;
  for (int e = blockIdx.x * blockDim.x + threadIdx.x; e < NEDGES; e += stride) {
    int d = (int)ei[NEDGES + e];  // edge_index[1]
    atomicAdd(&deg[d], 1.0f);
  }
}

__global__ void deg_rsqrt(float* deg) {
  int n = blockIdx.x * blockDim.x + threadIdx.x;
  if (n < NNODES) deg[n] = rsqrtf(deg[n]);  // deg >= 1 always
}

// ---------------- WMMA GEMM, 256 output cols ----------------
// A: [N x K] row-major, B: [K x 256] row-major, C: [N x 256].
// Block = 128 threads = 4 waves; block handles one 16-row tile; each wave a
// 16x64 strip (4 accumulators, A operand reused across the 4 WMMAs).
__global__ void gemm_wmma_n256(const float* __restrict__ A,
                               const float* __restrict__ B,
                               float* __restrict__ C, int K) {
  const int lane = threadIdx.x & 31;
  const int wave = threadIdx.x >> 5;   // 0..3
  const int h    = lane >> 4;          // half-wave selector
  const int m    = lane & 15;          // A-row / B-col / D-col index
  const int tile = blockIdx.x;
  const int colbase = wave * 64;
  const float* Arow = A + (size_t)(tile * 16 + m) * K;

  v8f acc0 = {}, acc1 = {}, acc2 = {}, acc3 = {};
  for (int k0 = 0; k0 < K; k0 += 4) {
    v2f a = *(const v2f*)(Arow + k0 + 2 * h);
    const float* B0 = B + (size_t)(k0 + 2 * h) * 256;      // K = 2h
    const float* B1 = B0 + 256;                            // K = 2h+1
    int c = colbase + m;
    v2f b;
    b.x = B0[c];      b.y = B1[c];      acc0 = wmma_f32(a, b, acc0);
    b.x = B0[c + 16]; b.y = B1[c + 16]; acc1 = wmma_f32(a, b, acc1);
    b.x = B0[c + 32]; b.y = B1[c + 32]; acc2 = wmma_f32(a, b, acc2);
    b.x = B0[c + 48]; b.y = B1[c + 48]; acc3 = wmma_f32(a, b, acc3);
  }
#pragma unroll
  for (int v = 0; v < 8; ++v) {
    size_t r = (size_t)(tile * 16 + v + 8 * h) * 256 + colbase + m;
    C[r]      = acc0[v];
    C[r + 16] = acc1[v];
    C[r + 32] = acc2[v];
    C[r + 48] = acc3[v];
  }
}

// ---------------- WMMA GEMM, 40 output cols (final layer) ----------------
// Each wave handles one 16-row tile, cols 0..39 (3 col-tiles, last partial).
__global__ void gemm_wmma_n40(const float* __restrict__ A,
                              const float* __restrict__ B,
                              float* __restrict__ C, int K) {
  const int lane = threadIdx.x & 31;
  const int wave = threadIdx.x >> 5;
  const int h = lane >> 4, m = lane & 15;
  const int tile = blockIdx.x * 4 + wave;   // uniform per wave
  if (tile >= NTILES) return;               // whole-wave exit: EXEC stays all-1s
  const float* Arow = A + (size_t)(tile * 16 + m) * K;

  v8f acc0 = {}, acc1 = {}, acc2 = {};
  const bool m8 = (m < 8);  // col 32+m valid only when m<8 (40 cols)
  for (int k0 = 0; k0 < K; k0 += 4) {
    v2f a = *(const v2f*)(Arow + k0 + 2 * h);
    const float* B0 = B + (size_t)(k0 + 2 * h) * DOUT;
    const float* B1 = B0 + DOUT;
    v2f b;
    b.x = B0[m];      b.y = B1[m];      acc0 = wmma_f32(a, b, acc0);
    b.x = B0[m + 16]; b.y = B1[m + 16]; acc1 = wmma_f32(a, b, acc1);
    b.x = m8 ? B0[m + 32] : 0.f;
    b.y = m8 ? B1[m + 32] : 0.f;
    acc2 = wmma_f32(a, b, acc2);
  }
#pragma unroll
  for (int v = 0; v < 8; ++v) {
    size_t r = (size_t)(tile * 16 + v + 8 * h) * DOUT + m;
    C[r]      = acc0[v];
    C[r + 16] = acc1[v];
    if (m8) C[r + 32] = acc2[v];
  }
}

// ---------------- self-loop init: agg = h * dinv^2 (overwrites agg) -------
__global__ void selfloop_init(const float* __restrict__ h,
                              const float* __restrict__ dinv,
                              float* __restrict__ agg, int F, int total) {
  int stride = gridDim.x * blockDim.x;
  for (int i = blockIdx.x * blockDim.x + threadIdx.x; i < total; i += stride) {
    int n = i / F;
    float dv = dinv[n];
    agg[i] = h[i] * dv * dv;
  }
}

// ---------------- edge scatter: agg[dst] += w * h[src] --------------------
__global__ void edge_scatter(const long long* __restrict__ ei,
                             const float* __restrict__ h,
                             const float* __restrict__ dinv,
                             float* __restrict__ agg, int F) {
  int nwaves = (gridDim.x * blockDim.x) >> 5;
  int wid = (blockIdx.x * blockDim.x + threadIdx.x) >> 5;
  int lane = threadIdx.x & 31;
  for (int e = wid; e < NEDGES; e += nwaves) {
    int s = (int)ei[e];
    int d = (int)ei[NEDGES + e];
    float w = dinv[s] * dinv[d];
    const float* hs = h + (size_t)s * F;
    float* ad = agg + (size_t)d * F;
    for (int f = lane; f < F; f += 32)
      atomicAdd(&ad[f], hs[f] * w);
  }
}

// ---------------- batchnorm (deterministic two-pass) ----------------------
__global__ void bn_partial(const float* __restrict__ agg,
                           const float* __restrict__ bias,
                           float* __restrict__ part) {
  int t = threadIdx.x;           // column 0..255
  float bb = bias[t];
  float s = 0.f, q = 0.f;
  for (int r = blockIdx.x; r < NNODES; r += NBN) {
    float v = agg[(size_t)r * 256 + t] + bb;
    s += v;
    q += v * v;
  }
  part[blockIdx.x * 512 + t] = s;
  part[blockIdx.x * 512 + 256 + t] = q;
}

__global__ void bn_finalize(const float* __restrict__ part,
                            const float* __restrict__ g,
                            const float* __restrict__ be,
                            float* __restrict__ scale,
                            float* __restrict__ shift) {
  int t = threadIdx.x;
  float s = 0.f, q = 0.f;
  for (int i = 0; i < NBN; ++i) {  // fixed order -> deterministic
    s += part[i * 512 + t];
    q += part[i * 512 + 256 + t];
  }
  float mu  = s / (float)NNODES;
  float var = q / (float)NNODES - mu * mu;
  float inv = rsqrtf(var + BN_EPS);
  float sc  = g[t] * inv;
  scale[t] = sc;
  shift[t] = be[t] - mu * sc;
}

__global__ void bn_apply_relu(float* __restrict__ agg,
                              const float* __restrict__ bias,
                              const float* __restrict__ scale,
                              const float* __restrict__ shift) {
  int stride = gridDim.x * blockDim.x;
  const int total = NNODES * 256;
  for (int i = blockIdx.x * blockDim.x + threadIdx.x; i < total; i += stride) {
    int f = i & 255;
    float v = (agg[i] + bias[f]) * scale[f] + shift[f];
    agg[i] = v > 0.f ? v : 0.f;
  }
}

// ---------------- log-softmax over 40 classes (wave per row) --------------
__global__ void logsoftmax40(float* __restrict__ out,
                             const float* __restrict__ bl) {
  int nwaves = (gridDim.x * blockDim.x) >> 5;
  int wid = (blockIdx.x * blockDim.x + threadIdx.x) >> 5;
  int lane = threadIdx.x & 31;
  bool has2 = (lane < DOUT - 32);  // lanes 0..7 own a second column
  for (int r = wid; r < NNODES; r += nwaves) {
    float* row = out + (size_t)r * DOUT;
    float v0 = row[lane] + bl[lane];
    float v1 = has2 ? row[lane + 32] + bl[lane + 32] : -3.4e38f;
    float mx = fmaxf(v0, v1);
#pragma unroll
    for (int o = 16; o > 0; o >>= 1) mx = fmaxf(mx, __shfl_xor(mx, o, 32));
    float s = expf(v0 - mx) + (has2 ? expf(v1 - mx) : 0.f);
#pragma unroll
    for (int o = 16; o > 0; o >>= 1) s += __shfl_xor(s, o, 32);
    float lg = logf(s) + mx;
    row[lane] = v0 - lg;
    if (has2) row[lane + 32] = v1 - lg;
  }
}

extern "C" void kernel_launch(void* const* d_in, const int* in_sizes, int n_in,
                              void* d_out, int out_size, void* d_ws,
                              size_t ws_size, hipStream_t stream) {
  (void)in_sizes; (void)n_in; (void)out_size; (void)ws_size;
  const float*     x   = (const float*)d_in[0];
  const long long* ei  = (const long long*)d_in[1];   // int64 [2, E]
  const float*     W1  = (const float*)d_in[2];
  const float*     b1  = (const float*)d_in[3];
  const float*     g1  = (const float*)d_in[4];
  const float*     be1 = (const float*)d_in[5];
  const float*     W2  = (const float*)d_in[6];
  const float*     b2  = (const float*)d_in[7];
  const float*     g2  = (const float*)d_in[8];
  const float*     be2 = (const float*)d_in[9];
  const float*     Wl  = (const float*)d_in[10];
  const float*     bl  = (const float*)d_in[11];
  float* out = (float*)d_out;

  // workspace carve-up (all regions fully overwritten every call)
  float* dinv  = (float*)d_ws;                         // NNODES (padded)
  float* Abuf  = dinv + 100352;                        // N x 256
  float* Bbuf  = Abuf + (size_t)NNODES * 256;          // N x 256
  float* part  = Bbuf + (size_t)NNODES * 256;          // NBN * 512
  float* scale = part + NBN * 512;                     // 256
  float* shift = scale + 256;                          // 256

  const int GS = 2048, BS = 256;

  // normalized adjacency weights
  init_deg<<<(NNODES + 255) / 256, 256, 0, stream>>>(dinv);
  deg_accum<<<GS, BS, 0, stream>>>(ei, dinv);
  deg_rsqrt<<<(NNODES + 255) / 256, 256, 0, stream>>>(dinv);

  // ---- layer 1: x @ W1 -> aggregate -> BN+ReLU ----
  gemm_wmma_n256<<<NTILES, 128, 0, stream>>>(x, W1, Abuf, DIN);
  selfloop_init<<<GS, BS, 0, stream>>>(Abuf, dinv, Bbuf, 256, NNODES * 256);
  edge_scatter<<<GS, BS, 0, stream>>>(ei, Abuf, dinv, Bbuf, 256);
  bn_partial<<<NBN, 256, 0, stream>>>(Bbuf, b1, part);
  bn_finalize<<<1, 256, 0, stream>>>(part, g1, be1, scale, shift);
  bn_apply_relu<<<GS, BS, 0, stream>>>(Bbuf, b1, scale, shift);

  // ---- layer 2 ----
  gemm_wmma_n256<<<NTILES, 128, 0, stream>>>(Bbuf, W2, Abuf, DH);
  selfloop_init<<<GS, BS, 0, stream>>>(Abuf, dinv, Bbuf, 256, NNODES * 256);
  edge_scatter<<<GS, BS, 0, stream>>>(ei, Abuf, dinv, Bbuf, 256);
  bn_partial<<<NBN, 256, 0, stream>>>(Bbuf, b2, part);
  bn_finalize<<<1, 256, 0, stream>>>(part, g2, be2, scale, shift);
  bn_apply_relu<<<GS, BS, 0, stream>>>(Bbuf, b2, scale, shift);

  // ---- layer 3: -> d_out, then log_softmax ----
  gemm_wmma_n40<<<(NTILES + 3) / 4, 128, 0, stream>>>(Bbuf, Wl, Abuf, DH);
  selfloop_init<<<GS, BS, 0, stream>>>(Abuf, dinv, out, DOUT, NNODES * DOUT);
  edge_scatter<<<GS, BS, 0, stream>>>(ei, Abuf, dinv, out, DOUT);
  logsoftmax40<<<GS, BS, 0, stream>>>(out, bl);
}